// PlanetoidGCN_73237782332060
// MI455X (gfx1250) — compile-verified
//
#include <hip/hip_runtime.h>
#include <hip/hip_bf16.h>
#include <math.h>

typedef __attribute__((ext_vector_type(2))) float v2f;
typedef __attribute__((ext_vector_type(8))) float v8f;

#define NFEAT 512
#define EMB   16
#define NCLS  40

// ---------------------------------------------------------------- utilities
__global__ __launch_bounds__(256) void fill_kernel(float* __restrict__ p, int n, float v) {
    int i = blockIdx.x * 256 + threadIdx.x;
    if (i < n) p[i] = v;
}

// deg[dst] += 1 per edge (deg pre-initialized to 1.0 for self loops)
__global__ __launch_bounds__(256) void edge_degree_kernel(const int* __restrict__ ei, int E,
                                                          float* __restrict__ deg) {
    int e = blockIdx.x * 256 + threadIdx.x;
    if (e < E) atomicAdd(&deg[ei[E + e]], 1.0f);
}

// deg -> rsqrt(max(deg,1)) in place
__global__ __launch_bounds__(256) void dinv_kernel(float* __restrict__ deg, int n) {
    int i = blockIdx.x * 256 + threadIdx.x;
    if (i < n) deg[i] = rsqrtf(fmaxf(deg[i], 1.0f));
}

// ---------------------------------------------------------------- GEMM 1: H = X @ W1  (fp32 WMMA)
__global__ __launch_bounds__(256) void gemm1_wmma_kernel(const float* __restrict__ X,
                                                         const float* __restrict__ W1,
                                                         float* __restrict__ H, int n_nodes) {
    __shared__ float Bs[NFEAT * EMB];           // 32 KB of 320 KB/WGP
    for (int i = threadIdx.x; i < NFEAT * EMB; i += 256) Bs[i] = W1[i];
    __syncthreads();

    const int wave = threadIdx.x >> 5;
    const int lane = threadIdx.x & 31;
    const int m0   = (blockIdx.x * 8 + wave) * 16;
    if (m0 >= n_nodes) return;                   // wave-uniform: EXEC stays all-1s

    const int col  = lane & 15;
    const int koff = lane >> 4;                  // 0: K={k,k+1}, 1: K={k+2,k+3}
    const float* __restrict__ arow = X + (size_t)(m0 + col) * NFEAT + 2 * koff;

    v8f acc = {};
    #pragma unroll 4
    for (int k = 0; k < NFEAT; k += 4) {
        v2f a = *(const v2f*)(arow + k);         // sequential along the row -> cache friendly
        v2f b;
        b.x = Bs[(k + 2 * koff + 0) * EMB + col];
        b.y = Bs[(k + 2 * koff + 1) * EMB + col];
        acc = __builtin_amdgcn_wmma_f32_16x16x4_f32(false, a, false, b, (short)0, acc,
                                                    false, false);
    }
    float* __restrict__ hout = H + (size_t)(m0 + 8 * koff) * EMB + col;
    #pragma unroll
    for (int v = 0; v < 8; ++v) hout[v * EMB] = acc[v];   // row m0 + 8*koff + v
}

// ---------------------------------------------------------------- scatter layer 1 (L2-resident atomics)
__global__ __launch_bounds__(256) void scatter1_kernel(const int* __restrict__ ei, int E,
                                                       const float* __restrict__ H,
                                                       const float* __restrict__ dinv,
                                                       float* __restrict__ AGG) {
    int tid = blockIdx.x * 256 + threadIdx.x;
    if (tid >= E * 4) return;
    int e = tid >> 2;
    int c = (tid & 3) << 2;
    int s = ei[e];
    int d = ei[E + e];
    float norm = dinv[s] * dinv[d];
    const float4 h = *(const float4*)(H + (size_t)s * EMB + c);
    float* a = AGG + (size_t)d * EMB + c;
    atomicAdd(a + 0, h.x * norm);
    atomicAdd(a + 1, h.y * norm);
    atomicAdd(a + 2, h.z * norm);
    atomicAdd(a + 3, h.w * norm);
}

// H1 = relu(AGG + H*dinv^2 + b1)   (self-loop folded in)
__global__ __launch_bounds__(256) void fuse1_kernel(const float* __restrict__ AGG,
                                                    const float* __restrict__ H,
                                                    const float* __restrict__ dinv,
                                                    const float* __restrict__ b1,
                                                    float* __restrict__ H1, int n_nodes) {
    int tid = blockIdx.x * 256 + threadIdx.x;
    if (tid >= n_nodes * EMB) return;
    int i = tid >> 4;
    int f = tid & 15;
    float di = dinv[i];
    float v  = AGG[tid] + H[tid] * di * di + b1[f];
    H1[tid]  = fmaxf(v, 0.0f);
}

// ---------------------------------------------------------------- GEMM 2: G = H1 @ W2 (fp32 WMMA, N padded 40->48)
__global__ __launch_bounds__(256) void gemm2_wmma_kernel(const float* __restrict__ H1,
                                                         const float* __restrict__ W2,
                                                         float* __restrict__ G, int n_nodes) {
    __shared__ float Bs[EMB * 48];
    for (int i = threadIdx.x; i < EMB * 48; i += 256) {
        int r = i / 48, c = i % 48;
        Bs[i] = (c < NCLS) ? W2[r * NCLS + c] : 0.0f;
    }
    __syncthreads();

    const int wave   = threadIdx.x >> 5;
    const int lane   = threadIdx.x & 31;
    const int mtiles = (n_nodes + 15) >> 4;
    const int tile   = blockIdx.x * 8 + wave;
    if (tile >= mtiles * 3) return;              // wave-uniform
    const int mt = tile / 3;
    const int nt = tile % 3;
    const int m0 = mt * 16;

    const int col  = lane & 15;
    const int koff = lane >> 4;
    const float* __restrict__ arow = H1 + (size_t)(m0 + col) * EMB + 2 * koff;

    v8f acc = {};
    #pragma unroll
    for (int k = 0; k < EMB; k += 4) {
        v2f a = { arow[k], arow[k + 1] };
        v2f b;
        b.x = Bs[(k + 2 * koff + 0) * 48 + nt * 16 + col];
        b.y = Bs[(k + 2 * koff + 1) * 48 + nt * 16 + col];
        acc = __builtin_amdgcn_wmma_f32_16x16x4_f32(false, a, false, b, (short)0, acc,
                                                    false, false);
    }
    int c = nt * 16 + col;
    if (c < NCLS) {
        float* __restrict__ gout = G + (size_t)(m0 + 8 * koff) * NCLS + c;
        #pragma unroll
        for (int v = 0; v < 8; ++v) gout[v * NCLS] = acc[v];
    }
}

// ---------------------------------------------------------------- scatter layer 2 into d_out
__global__ __launch_bounds__(256) void scatter2_kernel(const int* __restrict__ ei, int E,
                                                       const float* __restrict__ G,
                                                       const float* __restrict__ dinv,
                                                       float* __restrict__ OUT) {
    int tid = blockIdx.x * 256 + threadIdx.x;
    if (tid >= E * 10) return;
    int e = tid / 10;
    int c = (tid % 10) << 2;
    int s = ei[e];
    int d = ei[E + e];
    float norm = dinv[s] * dinv[d];
    const float4 g = *(const float4*)(G + (size_t)s * NCLS + c);
    float* a = OUT + (size_t)d * NCLS + c;
    atomicAdd(a + 0, g.x * norm);
    atomicAdd(a + 1, g.y * norm);
    atomicAdd(a + 2, g.z * norm);
    atomicAdd(a + 3, g.w * norm);
}

// logits = OUT + G*dinv^2 + b2; OUT = log_softmax(logits)   (one thread per node, all in registers)
__global__ __launch_bounds__(256) void final_kernel(float* __restrict__ OUT,
                                                    const float* __restrict__ G,
                                                    const float* __restrict__ dinv,
                                                    const float* __restrict__ b2, int n_nodes) {
    int i = blockIdx.x * 256 + threadIdx.x;
    if (i >= n_nodes) return;
    float di = dinv[i];
    float d2 = di * di;
    float v[NCLS];
    float mx = -INFINITY;
    #pragma unroll
    for (int f = 0; f < NCLS; ++f) {
        float t = OUT[(size_t)i * NCLS + f] + G[(size_t)i * NCLS + f] * d2 + b2[f];
        v[f] = t;
        mx = fmaxf(mx, t);
    }
    float s = 0.0f;
    #pragma unroll
    for (int f = 0; f < NCLS; ++f) s += __expf(v[f] - mx);
    float ls = __logf(s);
    #pragma unroll
    for (int f = 0; f < NCLS; ++f) OUT[(size_t)i * NCLS + f] = v[f] - mx - ls;
}

// ---------------------------------------------------------------- launch
extern "C" void kernel_launch(void* const* d_in, const int* in_sizes, int n_in,
                              void* d_out, int out_size, void* d_ws, size_t ws_size,
                              hipStream_t stream) {
    const float* x  = (const float*)d_in[0];
    const int*   ei = (const int*)d_in[1];
    const float* W1 = (const float*)d_in[2];
    const float* b1 = (const float*)d_in[3];
    const float* W2 = (const float*)d_in[4];
    const float* b2 = (const float*)d_in[5];
    float* out = (float*)d_out;

    const int N = in_sizes[0] / NFEAT;   // 100000
    const int E = in_sizes[1] / 2;       // 3200000

    float* ws   = (float*)d_ws;
    float* dinv = ws;                               // N
    float* H    = dinv + N;                         // N*16
    float* AGG  = H    + (size_t)N * EMB;           // N*16
    float* H1   = AGG  + (size_t)N * EMB;           // N*16
    float* G    = H1   + (size_t)N * EMB;           // N*40

    const int T = 256;
    // init: deg=1 (self loops), AGG=0, OUT(=layer-2 accumulator)=0
    fill_kernel<<<(N + T - 1) / T, T, 0, stream>>>(dinv, N, 1.0f);
    fill_kernel<<<(N * EMB + T - 1) / T, T, 0, stream>>>(AGG, N * EMB, 0.0f);
    fill_kernel<<<(N * NCLS + T - 1) / T, T, 0, stream>>>(out, N * NCLS, 0.0f);

    edge_degree_kernel<<<(E + T - 1) / T, T, 0, stream>>>(ei, E, dinv);
    dinv_kernel<<<(N + T - 1) / T, T, 0, stream>>>(dinv, N);

    // layer 1
    int mtiles = (N + 15) / 16;
    gemm1_wmma_kernel<<<(mtiles + 7) / 8, T, 0, stream>>>(x, W1, H, N);
    scatter1_kernel<<<(E * 4 + T - 1) / T, T, 0, stream>>>(ei, E, H, dinv, AGG);
    fuse1_kernel<<<(N * EMB + T - 1) / T, T, 0, stream>>>(AGG, H, dinv, b1, H1, N);

    // layer 2
    gemm2_wmma_kernel<<<(mtiles * 3 + 7) / 8, T, 0, stream>>>(H1, W2, G, N);
    scatter2_kernel<<<(E * 10 + T - 1) / T, T, 0, stream>>>(ei, E, G, dinv, out);

    // self-loop + bias + log_softmax
    final_kernel<<<(N + T - 1) / T, T, 0, stream>>>(out, G, dinv, b2, N);
}